// UIQAModule_33603824124082
// MI455X (gfx1250) — compile-verified
//
#include <hip/hip_runtime.h>
#include <math.h>

typedef __attribute__((ext_vector_type(2))) float v2f;
typedef __attribute__((ext_vector_type(8))) float v8f;

typedef const float __attribute__((address_space(1)))* gcf;
typedef float __attribute__((address_space(1)))* gf;
typedef const v2f __attribute__((address_space(1)))* gcv2;

#define TCH  512
#define MAXD 4096
#define NB   4

// ---------------------------------------------------------------------------
// fp32 WMMA GEMM tile body: wave computes 32(M) x (NS*16)(N), NS compile-time.
// A(m,k) at A[m*rsA + k*csA]; B(k,n) at B[k*rsB + n*csB]; C row-major (ldc).
// AC=true: csA==1, A-fragment pair loaded as one 8B-aligned b64.
// Branch-free inner loop: EXEC stays all-ones through every WMMA.
// ---------------------------------------------------------------------------
template<int NS, bool AC>
__device__ __forceinline__ void gemm_tile(
    gcf A, int rsA, int csA,
    gcf Bm, int rsB, int csB,
    gf C, int ldc,
    gcf bias, int bias_mode, float alpha,
    int m0, int n0, int K, int lane)
{
    int row   = lane & 15;           // M index in A frag / N index in B frag
    int khalf = (lane >> 4) << 1;    // lanes 16..31 hold K = 2,3 of each 4-step

    v8f acc0[NS], acc1[NS];
#pragma unroll
    for (int j = 0; j < NS; ++j) {
        acc0[j] = (v8f){0,0,0,0,0,0,0,0};
        acc1[j] = (v8f){0,0,0,0,0,0,0,0};
    }

    gcf Ap0 = A + (m0 + row) * rsA + khalf * csA;
    gcf Ap1 = Ap0 + 16 * rsA;
    gcf Bp[NS];
#pragma unroll
    for (int j = 0; j < NS; ++j)
        Bp[j] = Bm + (n0 + j * 16 + row) * csB + khalf * rsB;

#pragma unroll 2
    for (int k = 0; k < K; k += 4) {
        v2f a0, a1;
        if (AC) {
            a0 = *(gcv2)Ap0;                 // global_load_b64
            a1 = *(gcv2)Ap1;
        } else {
            a0.x = Ap0[0]; a0.y = Ap0[csA];
            a1.x = Ap1[0]; a1.y = Ap1[csA];
        }
        Ap0 += csA << 2;
        Ap1 += csA << 2;
#pragma unroll
        for (int j = 0; j < NS; ++j) {
            v2f bb;
            bb.x = Bp[j][0];
            bb.y = Bp[j][rsB];
            Bp[j] += rsB << 2;
            acc0[j] = __builtin_amdgcn_wmma_f32_16x16x4_f32(
                false, a0, false, bb, (short)0, acc0[j], false, false);
            acc1[j] = __builtin_amdgcn_wmma_f32_16x16x4_f32(
                false, a1, false, bb, (short)0, acc1[j], false, false);
        }
    }

    // C/D layout: VGPR r -> row m0 + r (+8 for lanes 16..31), col = lane&15
    int rbase = m0 + ((lane >> 4) << 3);
    int coff  = lane & 15;
#pragma unroll
    for (int j = 0; j < NS; ++j) {
        int col = n0 + j * 16 + coff;
        float bcol = (bias_mode == 2) ? bias[col] : 0.f;
#pragma unroll
        for (int r = 0; r < 8; ++r) {
            int rr = rbase + r;
            float v = acc0[j][r] * alpha + bcol;
            if (bias_mode == 1) v += bias[rr];
            C[rr * ldc + col] = v;
        }
#pragma unroll
        for (int r = 0; r < 8; ++r) {
            int rr = rbase + 16 + r;
            float v = acc1[j][r] * alpha + bcol;
            if (bias_mode == 1) v += bias[rr];
            C[rr * ldc + col] = v;
        }
    }
}

// Block = 256 thr = 8 waves arranged 2(M) x 4(N); wave tile 32 x 64.
// Block tile 64(M) x 256(N); partial N tiles dispatch to smaller NS bodies
// via wave-uniform scalar branches (readfirstlane-scalarized).
__global__ __launch_bounds__(256) void gemm_wmma_f32(
    const float* __restrict__ A_, int batchA, int rsA, int csA,
    const float* __restrict__ B_, int batchB, int rsB, int csB,
    float* __restrict__ C_, int batchC, int ldc,
    const float* __restrict__ bias_, int bias_mode,
    float alpha, int M, int N, int K)
{
    int bz = blockIdx.z;
    gcf A  = (gcf)A_  + bz * batchA;
    gcf Bm = (gcf)B_  + bz * batchB;
    gf  C  = (gf)C_   + bz * batchC;
    gcf bias = (gcf)bias_;

    int lane = threadIdx.x & 31;
    int wave = __builtin_amdgcn_readfirstlane((int)threadIdx.x >> 5); // SGPR
    int m0 = blockIdx.y * 64  + (wave >> 2) * 32;
    int n0 = blockIdx.x * 256 + (wave & 3) * 64;
    if (m0 >= M) return;                  // scalar branch, EXEC untouched
    int rem = (N - n0) >> 4;              // # of valid 16-wide subtiles
    if (rem <= 0) return;
    int nv = rem > 4 ? 4 : rem;

    if (csA == 1) {
        switch (nv) {
        case 4: gemm_tile<4, true>(A, rsA, csA, Bm, rsB, csB, C, ldc, bias, bias_mode, alpha, m0, n0, K, lane); break;
        case 3: gemm_tile<3, true>(A, rsA, csA, Bm, rsB, csB, C, ldc, bias, bias_mode, alpha, m0, n0, K, lane); break;
        case 2: gemm_tile<2, true>(A, rsA, csA, Bm, rsB, csB, C, ldc, bias, bias_mode, alpha, m0, n0, K, lane); break;
        default: gemm_tile<1, true>(A, rsA, csA, Bm, rsB, csB, C, ldc, bias, bias_mode, alpha, m0, n0, K, lane); break;
        }
    } else {
        switch (nv) {
        case 4: gemm_tile<4, false>(A, rsA, csA, Bm, rsB, csB, C, ldc, bias, bias_mode, alpha, m0, n0, K, lane); break;
        case 3: gemm_tile<3, false>(A, rsA, csA, Bm, rsB, csB, C, ldc, bias, bias_mode, alpha, m0, n0, K, lane); break;
        case 2: gemm_tile<2, false>(A, rsA, csA, Bm, rsB, csB, C, ldc, bias, bias_mode, alpha, m0, n0, K, lane); break;
        default: gemm_tile<1, false>(A, rsA, csA, Bm, rsB, csB, C, ldc, bias, bias_mode, alpha, m0, n0, K, lane); break;
        }
    }
}

// ---------------------------------------------------------------------------
// Strided 2x2 conv -> writes tokens into padded S [NB][512][4096]
// ---------------------------------------------------------------------------
__global__ __launch_bounds__(256) void conv2x2_kernel(
    const float* __restrict__ f, const float* __restrict__ cw,
    const float* __restrict__ cb, float* __restrict__ S,
    int C, int h, int w, int chOff, int d)
{
    int wf = w >> 1;
    long total = (long)NB * C * d;
    long idx = (long)blockIdx.x * 256 + threadIdx.x;
    if (idx >= total) return;
    int t  = (int)(idx % d);
    long r = idx / d;
    int co = (int)(r % C);
    int b  = (int)(r / C);
    int yf = t / wf, xf = t % wf;

    gcf fb = (gcf)f + (long)b * C * h * w + (long)(2 * yf) * w + 2 * xf;
    gcf wp = (gcf)cw + (long)co * C * 4;
    float acc = ((gcf)cb)[co];
    for (int ci = 0; ci < C; ++ci) {
        acc = fmaf(fb[0],     wp[0], acc);
        acc = fmaf(fb[1],     wp[1], acc);
        acc = fmaf(fb[w],     wp[2], acc);
        acc = fmaf(fb[w + 1], wp[3], acc);
        fb += h * w;
        wp += 4;
    }
    ((gf)S)[((long)b * TCH + chOff + co) * MAXD + t] = acc;
}

// ---------------------------------------------------------------------------
// Tiled transpose: out[c][r] = in[r][c]   (in: rows x cols)
// ---------------------------------------------------------------------------
__global__ __launch_bounds__(256) void transpose_kernel(
    const float* __restrict__ in, float* __restrict__ out, int rows, int cols)
{
    __shared__ float tile[32][33];
    int c0 = blockIdx.x * 32, r0 = blockIdx.y * 32;
    int tx = threadIdx.x & 31, ty = threadIdx.x >> 5;
    for (int i = ty; i < 32; i += 8) {
        int rr = r0 + i, cc = c0 + tx;
        tile[i][tx] = (rr < rows && cc < cols) ? ((gcf)in)[(long)rr * cols + cc] : 0.f;
    }
    __syncthreads();
    for (int i = ty; i < 32; i += 8) {
        int cc = c0 + i, rr = r0 + tx;
        if (cc < cols && rr < rows) ((gf)out)[(long)cc * rows + rr] = tile[tx][i];
    }
}

// ---------------------------------------------------------------------------
// Per-row max / sum / sumsq over one 4096-wide score row
// ---------------------------------------------------------------------------
__global__ __launch_bounds__(256) void row_stats_kernel(
    const float* __restrict__ sc, float* __restrict__ rowmax,
    float* __restrict__ rowsum, float* __restrict__ rowsq)
{
    __shared__ float sbuf[256];
    gcf p = (gcf)sc + (long)blockIdx.x * MAXD;
    float mx = -3.402823466e38f, sm = 0.f, sq = 0.f;
    for (int i = threadIdx.x; i < MAXD; i += 256) {
        float v = p[i];
        mx = fmaxf(mx, v);
        sm += v;
        sq = fmaf(v, v, sq);
    }
    int t = threadIdx.x;
    sbuf[t] = mx; __syncthreads();
    for (int s = 128; s > 0; s >>= 1) { if (t < s) sbuf[t] = fmaxf(sbuf[t], sbuf[t + s]); __syncthreads(); }
    float rmx = sbuf[0]; __syncthreads();
    sbuf[t] = sm; __syncthreads();
    for (int s = 128; s > 0; s >>= 1) { if (t < s) sbuf[t] += sbuf[t + s]; __syncthreads(); }
    float rsm = sbuf[0]; __syncthreads();
    sbuf[t] = sq; __syncthreads();
    for (int s = 128; s > 0; s >>= 1) { if (t < s) sbuf[t] += sbuf[t + s]; __syncthreads(); }
    if (t == 0) { rowmax[blockIdx.x] = rmx; rowsum[blockIdx.x] = rsm; rowsq[blockIdx.x] = sbuf[0]; }
}

// ---------------------------------------------------------------------------
// Reduce row sums -> mu, rstd for one batch element (deterministic)
// ---------------------------------------------------------------------------
__global__ __launch_bounds__(256) void batch_stats_kernel(
    const float* __restrict__ rowsum, const float* __restrict__ rowsq,
    int nrows, float* __restrict__ murstd)
{
    __shared__ float s1[256], s2[256];
    float a = 0.f, b = 0.f;
    for (int i = threadIdx.x; i < nrows; i += 256) { a += rowsum[i]; b += rowsq[i]; }
    int t = threadIdx.x;
    s1[t] = a; s2[t] = b; __syncthreads();
    for (int s = 128; s > 0; s >>= 1) {
        if (t < s) { s1[t] += s1[t + s]; s2[t] += s2[t + s]; }
        __syncthreads();
    }
    if (t == 0) {
        float total = (float)nrows * (float)MAXD;
        float mu  = s1[0] / total;
        float var = s2[0] / total - mu * mu;
        murstd[0] = mu;
        murstd[1] = rsqrtf(var + 1e-5f);
    }
}

// ---------------------------------------------------------------------------
// In-place instance-norm + softmax over each 4096-wide score row
// ---------------------------------------------------------------------------
__global__ __launch_bounds__(256) void softmax_kernel(
    float* __restrict__ sc, const float* __restrict__ rowmax,
    const float* __restrict__ murstd)
{
    __shared__ float sbuf[256];
    float mu = murstd[0], rstd = murstd[1];
    gf p = (gf)sc + (long)blockIdx.x * MAXD;
    float rm = (rowmax[blockIdx.x] - mu) * rstd;   // max of normalized row
    int t = threadIdx.x;
    float loc[16];
    float sm = 0.f;
#pragma unroll
    for (int j = 0; j < 16; ++j) {
        float v = (p[t + j * 256] - mu) * rstd - rm;
        v = expf(v);
        loc[j] = v;
        sm += v;
    }
    sbuf[t] = sm; __syncthreads();
    for (int s = 128; s > 0; s >>= 1) { if (t < s) sbuf[t] += sbuf[t + s]; __syncthreads(); }
    float inv = 1.f / sbuf[0];
#pragma unroll
    for (int j = 0; j < 16; ++j) p[t + j * 256] = loc[j] * inv;
}

// ---------------------------------------------------------------------------
// 2x bilinear upsample (half-pixel centers, edge clamp) + residual add
// ctx layout: [b][q = yf*wf+xf][c]
// ---------------------------------------------------------------------------
__global__ __launch_bounds__(256) void upsample_add_kernel(
    const float* __restrict__ fin, const float* __restrict__ ctx,
    float* __restrict__ outp, int C, int h, int w)
{
    int hf = h >> 1, wf = w >> 1;
    long total = (long)NB * C * h * w;
    long idx = (long)blockIdx.x * 256 + threadIdx.x;
    if (idx >= total) return;
    int x = (int)(idx % w);
    long r = idx / w;
    int y = (int)(r % h); r /= h;
    int c = (int)(r % C);
    int b = (int)(r / C);

    float sy = 0.5f * y - 0.25f;
    float sx = 0.5f * x - 0.25f;
    int y0 = (int)floorf(sy), x0 = (int)floorf(sx);
    float fy = sy - (float)y0, fx = sx - (float)x0;
    int y0c = min(max(y0, 0), hf - 1), y1c = min(max(y0 + 1, 0), hf - 1);
    int x0c = min(max(x0, 0), wf - 1), x1c = min(max(x0 + 1, 0), wf - 1);

    gcf cb = (gcf)ctx + ((long)b * hf * wf) * C + c;
    float v00 = cb[((long)y0c * wf + x0c) * C];
    float v01 = cb[((long)y0c * wf + x1c) * C];
    float v10 = cb[((long)y1c * wf + x0c) * C];
    float v11 = cb[((long)y1c * wf + x1c) * C];
    float v0 = v00 + fx * (v01 - v00);
    float v1 = v10 + fx * (v11 - v10);
    ((gf)outp)[idx] = fin[idx] + (v0 + fy * (v1 - v0));
}

// ---------------------------------------------------------------------------
extern "C" void kernel_launch(void* const* d_in, const int* in_sizes, int n_in,
                              void* d_out, int out_size, void* d_ws, size_t ws_size,
                              hipStream_t stream)
{
    static const int  Ca[4]     = {32, 64, 160, 256};
    static const int  Ha[4]     = {128, 64, 32, 16};
    static const int  Da[4]     = {4096, 1024, 256, 64};
    static const int  CHOFFa[4] = {0, 32, 96, 256};

    const float *f[4], *cw[4], *cbp[4], *qw[4], *qb[4];
    for (int i = 0; i < 4; ++i) {
        f[i]   = (const float*)d_in[i * 5 + 0];
        cw[i]  = (const float*)d_in[i * 5 + 1];
        cbp[i] = (const float*)d_in[i * 5 + 2];
        qw[i]  = (const float*)d_in[i * 5 + 3];
        qb[i]  = (const float*)d_in[i * 5 + 4];
    }
    const float* kw = (const float*)d_in[20];
    const float* kb = (const float*)d_in[21];
    const float* vw = (const float*)d_in[22];
    const float* vb = (const float*)d_in[23];

    float* ws = (float*)d_ws;
    long off = 0;
    float* S   = ws + off; off += (long)NB * TCH * MAXD;   // padded tokens [b][512][4096]
    float* Kc  = ws + off; off += (long)NB * TCH * MAXD;   // K channel-major [b][512][4096]
    float* Vt  = ws + off; off += (long)NB * MAXD * TCH;   // V token-major  [b][4096][512]
    float* Qb  = ws + off; off += (long)NB * MAXD * TCH;   // Q per scale    [b][d][512]
    float* Sc  = ws + off; off += (long)MAXD * MAXD;       // per-batch scores (L2-resident)
    float* Ctx = ws + off; off += (long)NB * MAXD * 256;   // context  [b][d][Ci]
    float* vwT = ws + off; off += (long)TCH * TCH;
    float* qwT = ws + off; off += (long)TCH * TCH;         // 4 transposed qw concatenated
    float* rowmax = ws + off; off += MAXD;
    float* rowsum = ws + off; off += MAXD;
    float* rowsq  = ws + off; off += MAXD;
    float* murstd = ws + off; off += 2;

    // zero S so padded tokens (t >= d_i) are 0
    hipMemsetAsync(S, 0, (size_t)NB * TCH * MAXD * sizeof(float), stream);

    // weight transposes (for coalesced B-fragments)
    {
        dim3 g(TCH / 32, TCH / 32);
        transpose_kernel<<<g, 256, 0, stream>>>(vw, vwT, TCH, TCH);
    }
    float* qwTp[4];
    long qoff = 0;
    for (int i = 0; i < 4; ++i) {
        qwTp[i] = qwT + qoff;
        dim3 g((Ca[i] + 31) / 32, TCH / 32);
        transpose_kernel<<<g, 256, 0, stream>>>(qw[i], qwTp[i], TCH, Ca[i]);
        qoff += (long)Ca[i] * TCH;
    }

    // strided convs -> S
    for (int i = 0; i < 4; ++i) {
        long total = (long)NB * Ca[i] * Da[i];
        conv2x2_kernel<<<(int)((total + 255) / 256), 256, 0, stream>>>(
            f[i], cw[i], cbp[i], S, Ca[i], Ha[i], Ha[i], CHOFFa[i], Da[i]);
    }

    // K (channel-major): Kc[b][c][t] = sum_u kw[c][u] * S[b][u][t] + kb[c]
    {
        dim3 g(MAXD / 256, TCH / 64, NB);
        gemm_wmma_f32<<<g, 256, 0, stream>>>(
            kw, 0, TCH, 1,
            S, TCH * MAXD, MAXD, 1,
            Kc, TCH * MAXD, MAXD,
            kb, 1, 1.0f, TCH, MAXD, TCH);
    }
    // V (token-major): Vt[b][t][c] = sum_u S[b][u][t] * vwT[u][c] + vb[c]
    {
        dim3 g(TCH / 256, MAXD / 64, NB);
        gemm_wmma_f32<<<g, 256, 0, stream>>>(
            S, TCH * MAXD, 1, MAXD,
            vwT, 0, TCH, 1,
            Vt, MAXD * TCH, TCH,
            vb, 2, 1.0f, MAXD, TCH, TCH);
    }

    const float invscale = 1.0f / (sqrtf((float)TCH) + 1e-6f);

    long outoff = 0;
    for (int i = 0; i < 4; ++i) {
        int Ci = Ca[i], d = Da[i];

        // Q[b][q][c] = sum_u S_i[b][u][q] * qwT_i[u][c] + qb[c]
        {
            dim3 g(TCH / 256, d / 64, NB);
            gemm_wmma_f32<<<g, 256, 0, stream>>>(
                S + (long)CHOFFa[i] * MAXD, TCH * MAXD, 1, MAXD,
                qwTp[i], 0, TCH, 1,
                Qb, d * TCH, TCH,
                qb[i], 2, 1.0f, d, TCH, Ci);
        }

        for (int b = 0; b < NB; ++b) {
            // scores[q][t] = (Q[b] @ Kc[b]) / scale   -- 64MB max, fits L2
            {
                dim3 g(MAXD / 256, d / 64, 1);
                gemm_wmma_f32<<<g, 256, 0, stream>>>(
                    Qb + (long)b * d * TCH, 0, TCH, 1,
                    Kc + (long)b * TCH * MAXD, 0, MAXD, 1,
                    Sc, 0, MAXD,
                    nullptr, 0, invscale, d, MAXD, TCH);
            }
            row_stats_kernel<<<d, 256, 0, stream>>>(Sc, rowmax, rowsum, rowsq);
            batch_stats_kernel<<<1, 256, 0, stream>>>(rowsum, rowsq, d, murstd);
            softmax_kernel<<<d, 256, 0, stream>>>(Sc, rowmax, murstd);
            // ctx[q][c] = probs @ Vt[b][:, :Ci]
            {
                dim3 g((Ci + 255) / 256, d / 64, 1);
                gemm_wmma_f32<<<g, 256, 0, stream>>>(
                    Sc, 0, MAXD, 1,
                    Vt + (long)b * MAXD * TCH, 0, TCH, 1,
                    Ctx + (long)b * d * Ci, 0, Ci,
                    nullptr, 0, 1.0f, d, Ci, MAXD);
            }
        }

        // bilinear 2x upsample + residual
        {
            int h = Ha[i], w = Ha[i];
            long total = (long)NB * Ci * h * w;
            upsample_add_kernel<<<(int)((total + 255) / 256), 256, 0, stream>>>(
                f[i], Ctx, (float*)d_out + outoff, Ci, h, w);
            outoff += total;
        }
    }
}